// MLP_45217415692507
// MI455X (gfx1250) — compile-verified
//
#include <hip/hip_runtime.h>
#include <math.h>

// ---------------------------------------------------------------------------
// HDRNet-style fused bilateral-grid pipeline for MI455X (gfx1250, wave32).
//   stage1: cm1 = tanh(W1b·relu(W1a·src+b1a)+b1b)        [4 guide ch]
//           hidden = relu(apply(slice(grid1, cm1), src))  [8 ch]
//   stage2: cm2 = tanh(W2b·relu(W2a·hidden+b2a)+b2b)     [9 guide ch]
//           out = apply(slice(grid2, cm2), hidden)        [3 ch]
// Conv1x1 layers run on V_WMMA_F32_16X16X4_F32 (full f32, K=4 chained).
// Cross-half lane exchange: v_permlane16_swap_b32 (1 VALU op, yields BOTH
// B-fragment patterns at once); ds_swizzle fallback if builtin missing.
// Grid copies use GLOBAL_LOAD_ASYNC_TO_LDS_B128 (ASYNCcnt) overlapped with
// the LDS-free guide-MLP phases; synchronous fallback if builtin missing.
// Slicing: 8 precomputed corner weights (apply-scale folded) -> 8 LDS FMAs
// per sliced channel accumulating directly into the output.
// ---------------------------------------------------------------------------

static constexpr int HH = 1024;
static constexpr int WW = 1024;
static constexpr int DG = 8;    // grid depth
static constexpr int GHc = 16;  // grid height
static constexpr int GWc = 16;  // grid width
static constexpr int VOL = DG * GHc * GWc;  // 2048 floats per grid channel

typedef float v2f __attribute__((ext_vector_type(2)));
typedef float v8f __attribute__((ext_vector_type(8)));
typedef unsigned int uv2 __attribute__((ext_vector_type(2)));
typedef int v4i __attribute__((vector_size(16)));

#if defined(__has_builtin)
#if __has_builtin(__builtin_amdgcn_permlane16_swap)
#define HAVE_PLSWAP 1
#endif
#if __has_builtin(__builtin_amdgcn_global_load_async_to_lds_b128)
#define HAVE_ASYNC 1
#endif
#endif

// D = A(16x4 f32) * B(4x16 f32) + C(16x16 f32); one wave per 16x16 tile.
__device__ __forceinline__ v8f wmma4(v2f a, v2f b, v8f c) {
  return __builtin_amdgcn_wmma_f32_16x16x4_f32(
      /*neg_a=*/false, a, /*neg_b=*/false, b,
      /*c_mod=*/(short)0, c, /*reuse_a=*/false, /*reuse_b=*/false);
}

// Lane swap with partner lane^16 (fallback path): ds_swizzle SWAPX16.
__device__ __forceinline__ float swap16(float x) {
  return __int_as_float(__builtin_amdgcn_ds_swizzle(__float_as_int(x), 0x401F));
}

// Half-wave swap of two registers:
//   a' = {a[0..15], b[0..15]}   (pattern A: low half own, high half partner-low)
//   b' = {a[16..31], b[16..31]} (pattern B: low half partner-high, high half own)
// On CDNA5 this is a single v_permlane16_swap_b32.
__device__ __forceinline__ void halfswap(float& a, float& b, bool lo) {
#if defined(HAVE_PLSWAP)
  uv2 r = __builtin_amdgcn_permlane16_swap(__float_as_uint(a), __float_as_uint(b),
                                           false, false);
  a = __uint_as_float(r.x);
  b = __uint_as_float(r.y);
#else
  float sa = swap16(a), sb = swap16(b);
  float na = lo ? a : sb;
  float nb = lo ? sa : b;
  a = na;
  b = nb;
#endif
}

// A fragment: 16x4 K-chunk [kb..kb+3] of weight matrix W[Co][Ci] (row-major),
// zero-padded outside. Lane m%16 holds row M=m; VGPR0/1 = K pair,
// lanes 0-15 -> K=kb..kb+1, lanes 16-31 -> K=kb+2..kb+3.
__device__ __forceinline__ v2f afrag(const float* __restrict__ Wt,
                                     int Co, int Ci, int kb, int lane) {
  int m = lane & 15;
  int k0 = kb + ((lane & 16) ? 2 : 0);
  v2f a;
  a.x = (m < Co && (k0    ) < Ci) ? Wt[m * Ci + k0    ] : 0.0f;
  a.y = (m < Co && (k0 + 1) < Ci) ? Wt[m * Ci + k0 + 1] : 0.0f;
  return a;
}

// C fragment carrying the bias, broadcast across columns.
__device__ __forceinline__ v8f cfrag(const float* __restrict__ b, int Co, int lane) {
  int moff = (lane & 16) ? 8 : 0;
  v8f c;
#pragma unroll
  for (int j = 0; j < 8; ++j) {
    int m = j + moff;
    c[j] = (m < Co) ? b[m] : 0.0f;
  }
  return c;
}

__device__ __forceinline__ v8f vrelu(v8f x) {
  v8f r;
#pragma unroll
  for (int j = 0; j < 8; ++j) r[j] = fmaxf(x[j], 0.0f);
  return r;
}

// Full K=16 layer on one tile: input is this tile's own (relu'd) D fragment.
// Channels 0-7 on low lanes (regs 0-7), channels 8-15 on high lanes.
// halfswap produces pattern-A (chunks K=0,4) and pattern-B (chunks K=8,12)
// fragments simultaneously.
__device__ __forceinline__ v8f mlp16(v8f r, const v2f* A, v8f acc, bool lo) {
  float q0 = r[0], q1 = r[1], q2 = r[2], q3 = r[3];
  float q4 = r[4], q5 = r[5], q6 = r[6], q7 = r[7];
  halfswap(q0, q2, lo);
  halfswap(q1, q3, lo);
  halfswap(q4, q6, lo);
  halfswap(q5, q7, lo);
  v2f b;
  b.x = q0; b.y = q1; acc = wmma4(A[0], b, acc);  // K chunk 0..3
  b.x = q4; b.y = q5; acc = wmma4(A[1], b, acc);  // K chunk 4..7
  b.x = q2; b.y = q3; acc = wmma4(A[2], b, acc);  // K chunk 8..11
  b.x = q6; b.y = q7; acc = wmma4(A[3], b, acc);  // K chunk 12..15
  return acc;
}

// Redistribute guide-MLP outputs to per-lane pixels.
__device__ __forceinline__ float pickA(float a, float b, bool lo) {
  halfswap(a, b, lo);
  return a;  // {tileA low lanes, tileB low lanes}
}
__device__ __forceinline__ float pickB(float a, float b, bool lo) {
  halfswap(a, b, lo);
  return b;  // {tileA high lanes, tileB high lanes}
}

__device__ __forceinline__ void zparams(float cm, int& z0, int& z1, float& wz) {
  float iz = (cm + 1.0f) * (0.5f * (float)DG) - 0.5f;
  iz = fminf(fmaxf(iz, 0.0f), (float)(DG - 1));
  z0 = (int)iz;
  wz = iz - (float)z0;
  z1 = (z0 + 1 < DG - 1) ? (z0 + 1) : (DG - 1);
}

// 8-corner weighted gather from an LDS-resident grid channel.
__device__ __forceinline__ float slice8(const float* __restrict__ v,
                                        int zb0, int zb1,
                                        int p00, int p01, int p10, int p11,
                                        float w000, float w001, float w010, float w011,
                                        float w100, float w101, float w110, float w111,
                                        float acc) {
  acc = fmaf(v[zb0 + p00], w000, acc);
  acc = fmaf(v[zb0 + p01], w001, acc);
  acc = fmaf(v[zb0 + p10], w010, acc);
  acc = fmaf(v[zb0 + p11], w011, acc);
  acc = fmaf(v[zb1 + p00], w100, acc);
  acc = fmaf(v[zb1 + p01], w101, acc);
  acc = fmaf(v[zb1 + p10], w110, acc);
  acc = fmaf(v[zb1 + p11], w111, acc);
  return acc;
}

// Issue a global->LDS copy of n4 float4s. Async path uses
// GLOBAL_LOAD_ASYNC_TO_LDS_B128 (no VGPR staging, tracked by ASYNCcnt) so the
// caller can overlap compute and wait later; fallback is a blocking copy.
__device__ __forceinline__ void issue_copy_f4(const float* __restrict__ gsrc,
                                              float* ldst, int n4, int tid) {
#if defined(HAVE_ASYNC)
  float4* gs = (float4*)gsrc;  // builtin takes non-const pointers
  float4* gd = (float4*)ldst;
  for (int i = tid; i < n4; i += 256) {
    __builtin_amdgcn_global_load_async_to_lds_b128(
        (__attribute__((address_space(1))) v4i*)(gs + i),
        (__attribute__((address_space(3))) v4i*)(gd + i),
        /*offset=*/0, /*cpol=*/0);
  }
#else
  const float4* gs = (const float4*)gsrc;
  float4* gd = (float4*)ldst;
  for (int i = tid; i < n4; i += 256) gd[i] = gs[i];
#endif
}

__device__ __forceinline__ void wait_copy() {
#if defined(HAVE_ASYNC)
#if defined(__has_builtin) && __has_builtin(__builtin_amdgcn_s_wait_asynccnt)
  __builtin_amdgcn_s_wait_asynccnt(0);
#else
  asm volatile("s_wait_asynccnt 0" ::: "memory");
#endif
#endif
}

__global__ __launch_bounds__(256) void hdrnet_fused(
    const float* __restrict__ src,
    const float* __restrict__ grid1, const float* __restrict__ grid2,
    const float* __restrict__ w1a, const float* __restrict__ b1a,
    const float* __restrict__ w1b, const float* __restrict__ b1b,
    const float* __restrict__ w2a, const float* __restrict__ b2a,
    const float* __restrict__ w2b, const float* __restrict__ b2b,
    float* __restrict__ out) {
  extern __shared__ float lds[];  // 256KB: grid1 first, then reused for grid2

  const int tid = threadIdx.x;
  const int lane = tid & 31;
  const int wave = tid >> 5;
  const bool lo = (lane < 16);

  // ==== phase A: launch grid1 DMA, then LDS-free stage-1 guide MLP ====
  issue_copy_f4(grid1, lds, (32 * VOL) / 4, tid);

  // Weight fragments (loop invariant).
  v2f A1 = afrag(w1a, 16, 3, 0, lane);  // 3->16, single K chunk (pad K=3)
  v8f C1 = cfrag(b1a, 16, lane);
  v2f A2[4];
#pragma unroll
  for (int q = 0; q < 4; ++q) A2[q] = afrag(w1b, 4, 16, 4 * q, lane);  // 16->4
  v8f C2 = cfrag(b1b, 4, lane);
  v2f A3[2];
#pragma unroll
  for (int q = 0; q < 2; ++q) A3[q] = afrag(w2a, 16, 8, 4 * q, lane);  // 8->16
  v8f C3 = cfrag(b2a, 16, lane);
  v2f A4[4];
#pragma unroll
  for (int q = 0; q < 4; ++q) A4[q] = afrag(w2b, 9, 16, 4 * q, lane);  // 16->9
  v8f C4 = cfrag(b2b, 9, lane);

  // Tile: 32x32 pixels per block; wave w owns rows 4w..4w+3; lane -> x.
  const int bx = (blockIdx.x & 31) << 5;
  const int by = (blockIdx.x >> 5) << 5;
  const int x = bx + lane;

  float s[4][3];    // per-chunk src pixels
  float cm1[4][4];  // stage-1 guide values

#pragma unroll
  for (int chk = 0; chk < 4; ++chk) {
    const int y = by + (wave << 2) + chk;
    const int pix = y * WW + x;
    s[chk][0] = src[pix];
    s[chk][1] = src[HH * WW + pix];
    s[chk][2] = src[2 * HH * WW + pix];

    // Layer 1 (3->16): one halfswap pair builds BOTH tiles' B fragments.
    float u0 = s[chk][0], u1 = s[chk][1], u2 = s[chk][2], uz = 0.0f;
    halfswap(u0, u2, lo);  // u0 = tileA.x, u2 = tileB.x
    halfswap(u1, uz, lo);  // u1 = tileA.y, uz = tileB.y (zero row preserved)
    v2f bA, bB;
    bA.x = u0; bA.y = u1;
    bB.x = u2; bB.y = uz;
    v8f r1a = vrelu(wmma4(A1, bA, C1));
    v8f r1b = vrelu(wmma4(A1, bB, C1));
    // Layer 2 (16->4), chained K=4 WMMAs.
    v8f d2a = mlp16(r1a, A2, C2, lo);
    v8f d2b = mlp16(r1b, A2, C2, lo);
#pragma unroll
    for (int g = 0; g < 4; ++g) cm1[chk][g] = tanhf(pickA(d2a[g], d2b[g], lo));
  }

  wait_copy();
  __syncthreads();  // grid1 resident in LDS

  // Spatial interp params in x (shared by both stages).
  float ix = (float)x * ((float)GWc / (float)WW) - 0.5f;
  ix = fminf(fmaxf(ix, 0.0f), (float)(GWc - 1));
  int x0 = (int)ix;
  float wxf = ix - (float)x0;
  int x1 = (x0 + 1 < GWc - 1) ? (x0 + 1) : (GWc - 1);

  float hid[4][8];  // per-lane hidden activations

  // ==== phase B: slice grid1 + apply + relu ====
#pragma unroll
  for (int chk = 0; chk < 4; ++chk) {
    const int y = by + (wave << 2) + chk;
    float iy = (float)y * ((float)GHc / (float)HH) - 0.5f;
    iy = fminf(fmaxf(iy, 0.0f), (float)(GHc - 1));
    int y0 = (int)iy;
    float wyf = iy - (float)y0;
    int y1 = (y0 + 1 < GHc - 1) ? (y0 + 1) : (GHc - 1);

    const int p00 = y0 * GWc + x0, p01 = y0 * GWc + x1;
    const int p10 = y1 * GWc + x0, p11 = y1 * GWc + x1;
    const float a00 = (1.0f - wxf) * (1.0f - wyf);
    const float a01 = wxf * (1.0f - wyf);
    const float a10 = (1.0f - wxf) * wyf;
    const float a11 = wxf * wyf;

    float h[8];
#pragma unroll
    for (int o = 0; o < 8; ++o) h[o] = 0.0f;
#pragma unroll
    for (int g = 0; g < 4; ++g) {
      int z0, z1;
      float wz;
      zparams(cm1[chk][g], z0, z1, wz);
      const int zb0 = z0 * (GHc * GWc), zb1 = z1 * (GHc * GWc);
      const float sg = (g < 3) ? s[chk][g] : 1.0f;
      const float wz0 = (1.0f - wz) * sg, wz1 = wz * sg;
      const float w000 = a00 * wz0, w001 = a01 * wz0, w010 = a10 * wz0, w011 = a11 * wz0;
      const float w100 = a00 * wz1, w101 = a01 * wz1, w110 = a10 * wz1, w111 = a11 * wz1;
#pragma unroll
      for (int o = 0; o < 8; ++o) {
        h[o] = slice8(lds + (g * 8 + o) * VOL, zb0, zb1, p00, p01, p10, p11,
                      w000, w001, w010, w011, w100, w101, w110, w111, h[o]);
      }
    }
#pragma unroll
    for (int o = 0; o < 8; ++o) hid[chk][o] = fmaxf(h[o], 0.0f);
  }

  __syncthreads();  // all grid1 reads complete

  // ==== phase C: launch grid2 DMA, then LDS-free stage-2 guide MLP ====
  issue_copy_f4(grid2, lds, (27 * VOL) / 4, tid);

  float cm2[4][9];
#pragma unroll
  for (int chk = 0; chk < 4; ++chk) {
    float t0 = hid[chk][0], t1 = hid[chk][1], t2 = hid[chk][2], t3 = hid[chk][3];
    float t4 = hid[chk][4], t5 = hid[chk][5], t6 = hid[chk][6], t7 = hid[chk][7];
    halfswap(t0, t2, lo);
    halfswap(t1, t3, lo);
    halfswap(t4, t6, lo);
    halfswap(t5, t7, lo);
    v2f b;
    v8f e1a = C3;
    b.x = t0; b.y = t1; e1a = wmma4(A3[0], b, e1a);
    b.x = t4; b.y = t5; e1a = wmma4(A3[1], b, e1a);
    v8f e1b = C3;
    b.x = t2; b.y = t3; e1b = wmma4(A3[0], b, e1b);
    b.x = t6; b.y = t7; e1b = wmma4(A3[1], b, e1b);
    v8f r2a = vrelu(e1a);
    v8f r2b = vrelu(e1b);
    v8f f2a = mlp16(r2a, A4, C4, lo);
    v8f f2b = mlp16(r2b, A4, C4, lo);
#pragma unroll
    for (int g = 0; g < 8; ++g) cm2[chk][g] = tanhf(pickA(f2a[g], f2b[g], lo));
    cm2[chk][8] = tanhf(pickB(f2a[0], f2b[0], lo));  // ch 8: high half, reg 0
  }

  wait_copy();
  __syncthreads();  // grid2 resident in LDS

  // ==== phase D: slice grid2 + apply + store ====
#pragma unroll
  for (int chk = 0; chk < 4; ++chk) {
    const int y = by + (wave << 2) + chk;
    float iy = (float)y * ((float)GHc / (float)HH) - 0.5f;
    iy = fminf(fmaxf(iy, 0.0f), (float)(GHc - 1));
    int y0 = (int)iy;
    float wyf = iy - (float)y0;
    int y1 = (y0 + 1 < GHc - 1) ? (y0 + 1) : (GHc - 1);

    const int p00 = y0 * GWc + x0, p01 = y0 * GWc + x1;
    const int p10 = y1 * GWc + x0, p11 = y1 * GWc + x1;
    const float a00 = (1.0f - wxf) * (1.0f - wyf);
    const float a01 = wxf * (1.0f - wyf);
    const float a10 = (1.0f - wxf) * wyf;
    const float a11 = wxf * wyf;

    float o0 = 0.0f, o1 = 0.0f, o2 = 0.0f;
#pragma unroll
    for (int k = 0; k < 9; ++k) {
      int z0, z1;
      float wz;
      zparams(cm2[chk][k], z0, z1, wz);
      const int zb0 = z0 * (GHc * GWc), zb1 = z1 * (GHc * GWc);
      const float hk = (k < 8) ? hid[chk][k] : 1.0f;
      const float wz0 = (1.0f - wz) * hk, wz1 = wz * hk;
      const float w000 = a00 * wz0, w001 = a01 * wz0, w010 = a10 * wz0, w011 = a11 * wz0;
      const float w100 = a00 * wz1, w101 = a01 * wz1, w110 = a10 * wz1, w111 = a11 * wz1;
      o0 = slice8(lds + (k * 3 + 0) * VOL, zb0, zb1, p00, p01, p10, p11,
                  w000, w001, w010, w011, w100, w101, w110, w111, o0);
      o1 = slice8(lds + (k * 3 + 1) * VOL, zb0, zb1, p00, p01, p10, p11,
                  w000, w001, w010, w011, w100, w101, w110, w111, o1);
      o2 = slice8(lds + (k * 3 + 2) * VOL, zb0, zb1, p00, p01, p10, p11,
                  w000, w001, w010, w011, w100, w101, w110, w111, o2);
    }

    const int pix = y * WW + x;
    out[pix] = o0;
    out[HH * WW + pix] = o1;
    out[2 * HH * WW + pix] = o2;
  }
}

extern "C" void kernel_launch(void* const* d_in, const int* in_sizes, int n_in,
                              void* d_out, int out_size, void* d_ws, size_t ws_size,
                              hipStream_t stream) {
  const float* src   = (const float*)d_in[0];
  const float* grid1 = (const float*)d_in[1];
  const float* grid2 = (const float*)d_in[2];
  const float* w1a   = (const float*)d_in[3];
  const float* b1a   = (const float*)d_in[4];
  const float* w1b   = (const float*)d_in[5];
  const float* b1b   = (const float*)d_in[6];
  const float* w2a   = (const float*)d_in[7];
  const float* b2a   = (const float*)d_in[8];
  const float* w2b   = (const float*)d_in[9];
  const float* b2b   = (const float*)d_in[10];
  float* out = (float*)d_out;

  (void)in_sizes; (void)n_in; (void)out_size; (void)d_ws; (void)ws_size;

  // 1024x1024 image, 32x32-pixel tiles -> 1024 blocks of 256 threads.
  // Dynamic LDS: 32 channels * 2048 floats * 4B = 256KB (fits 320KB/WGP;
  // grid2's 216KB reuses the same buffer after the mid-kernel barrier).
  dim3 grid(1024), block(256);
  size_t shmem = (size_t)(32 * VOL) * sizeof(float);
  hipLaunchKernelGGL(hdrnet_fused, grid, block, shmem, stream,
                     src, grid1, grid2, w1a, b1a, w1b, b1b,
                     w2a, b2a, w2b, b2b, out);
}